// ARMAMultiHeadAttention_66400194396830
// MI455X (gfx1250) — compile-verified
//
#include <hip/hip_runtime.h>
#include <hip/hip_bf16.h>

#define Bb 2
#define Ss 2048
#define Dd 1024
#define Hh 16
#define DEPTH 64

typedef __attribute__((ext_vector_type(2))) float v2f;
typedef __attribute__((ext_vector_type(8))) float v8f;
typedef __attribute__((ext_vector_type(4))) unsigned int v4u;
typedef __attribute__((ext_vector_type(8))) int v8i;
typedef __attribute__((ext_vector_type(4))) int v4i;

__device__ __forceinline__ v8f wmma_f32(v2f a, v2f b, v8f c) {
  return __builtin_amdgcn_wmma_f32_16x16x4_f32(false, a, false, b, (short)0, c,
                                               false, false);
}

// TDM: stage a 16x64 f32 tile (row stride Dd) into LDS with +1-DWORD row pad
// (LDS row stride 65 floats -> bank-conflict-free column reads).
// D# per CDNA5 ISA ch.8: group0 = count/lds_addr/global_addr/type,
// group1 = dims/strides/pad, groups 2-3 unused (2D tensor).
__device__ __forceinline__ void tdm_load_k16x64(const float* gptr,
                                                unsigned lds_off) {
  unsigned long long ga = (unsigned long long)(size_t)gptr;
  v4u g0;
  g0.x = 1u;                                   // count=1 (valid user D#)
  g0.y = lds_off;                              // lds_addr (bytes)
  g0.z = (unsigned)(ga & 0xFFFFFFFFu);         // global_addr[31:0]
  g0.w = ((unsigned)(ga >> 32) & 0x01FFFFFFu)  // global_addr[56:32]
         | 0x80000000u;                        // type=2 ("image")
  v8i g1;
  g1[0] = 0x01520000;  // data_size=4B | pad_enable | pad_interval=64dw (code 5)
  g1[1] = 0x00400000;  // tensor_dim0 = 64   (bits 79:48)
  g1[2] = 0x00100000;  // tensor_dim1 = 16   (bits 111:80)
  g1[3] = 0x00400000;  // tile_dim0  = 64    (bits 127:112)
  g1[4] = 16;          // tile_dim1  = 16
  g1[5] = Dd;          // tensor_dim0_stride = 1024 elements
  g1[6] = 0;
  g1[7] = 0;
  v4i z4 = {0, 0, 0, 0};
  v8i z8 = {0, 0, 0, 0, 0, 0, 0, 0};
  __builtin_amdgcn_tensor_load_to_lds(g0, g1, z4, z4, z8, 0);
}

// C[M, n0:n0+64] = A[M,K] @ W[K, n0:n0+64] + bias ; one wave per 16x64 strip,
// A-fragment reused across 4 WMMAs per K-step.
__global__ __launch_bounds__(32) void gemm16x64(const float* __restrict__ A,
                                                const float* __restrict__ W,
                                                const float* __restrict__ bias,
                                                float* __restrict__ C,
                                                int M, int N, int K) {
  const int lane = threadIdx.x & 31;
  const int half = lane >> 4;  // selects K pair {0,1} vs {2,3}
  const int l16 = lane & 15;
  const int n0 = blockIdx.x * 64;
  const int m0 = blockIdx.y * 16;
  v8f acc[4] = {v8f{}, v8f{}, v8f{}, v8f{}};
  const float* Arow = A + (size_t)(m0 + l16) * K;
  for (int k = 0; k < K; k += 4) {
    v2f a;
    a.x = Arow[k + 2 * half];
    a.y = Arow[k + 2 * half + 1];
    const float* w0 = W + (size_t)(k + 2 * half) * N + n0 + l16;
#pragma unroll
    for (int nt = 0; nt < 4; ++nt) {
      v2f b;
      b.x = w0[nt * 16];
      b.y = w0[nt * 16 + N];
      acc[nt] = wmma_f32(a, b, acc[nt]);
    }
  }
#pragma unroll
  for (int nt = 0; nt < 4; ++nt) {
    const float bv = bias[n0 + nt * 16 + l16];
#pragma unroll
    for (int r = 0; r < 8; ++r)
      C[(size_t)(m0 + r + 8 * half) * N + n0 + nt * 16 + l16] = acc[nt][r] + bv;
  }
}

// rs[b,h,t] = sum_d Vp[b,t,h*64+d]   (per-head rowsum feeding AR/MA bias)
__global__ __launch_bounds__(512) void rowsum_heads(const float* __restrict__ Vp,
                                                    float* __restrict__ rs) {
  const int bt = blockIdx.x;  // b*S + t
  const int h = threadIdx.x >> 5;
  const int lane = threadIdx.x & 31;
  const float* p = Vp + (size_t)bt * Dd + h * DEPTH;
  float v = p[lane] + p[lane + 32];
#pragma unroll
  for (int m = 16; m >= 1; m >>= 1) v += __shfl_xor(v, m, 32);
  if (lane == 0) {
    const int b = bt / Ss, t = bt % Ss;
    rs[((size_t)b * Hh + h) * Ss + t] = v;
  }
}

// Fused ARMA attention, one wave per (b,h, 16 query rows).
// K tiles are TDM-staged into double-buffered LDS (tensor_load_to_lds +
// s_wait_tensorcnt). Pass 1: online softmax stats. Pass 2: recompute logits,
// write attn exactly once, route P through LDS to A-layout, ctx += P @ V.
__global__ __launch_bounds__(32) void arma_attn(
    const float* __restrict__ Qp, const float* __restrict__ Kp,
    const float* __restrict__ Vp, const float* __restrict__ mask,
    const float* __restrict__ rs, const float* __restrict__ arw,
    const float* __restrict__ maw, const float* __restrict__ alphap,
    const float* __restrict__ betap, float* __restrict__ attn,
    float* __restrict__ ctx) {
  __shared__ float kbuf[2][16 * 65];  // TDM-padded K tiles (row stride 65)
  __shared__ float pT[16 * 16];
  const int lane = threadIdx.x & 31;
  const int half = lane >> 4;
  const int l16 = lane & 15;
  const int q0 = blockIdx.x * 16;
  const int bh = blockIdx.y;  // b*H + h
  const int b = bh / Hh, h = bh % Hh;
  const float scale = 0.125f;  // 1/sqrt(64)
  const float alpha = alphap[0], beta = betap[0], aw = arw[0], mw = maw[0];

  const float* qbase = Qp + (size_t)b * Ss * Dd + (size_t)h * DEPTH;
  const float* kbase = Kp + (size_t)b * Ss * Dd + (size_t)h * DEPTH;
  const float* vbase = Vp + (size_t)b * Ss * Dd + (size_t)h * DEPTH;
  const float* rsb = rs + (size_t)bh * Ss;
  const unsigned kb_off[2] = {(unsigned)(size_t)&kbuf[0][0],
                              (unsigned)(size_t)&kbuf[1][0]};

  float qbias[8], mrow[8], lrow[8];
#pragma unroll
  for (int r = 0; r < 8; ++r) {
    const int q = q0 + r + 8 * half;
    float bias = 0.f;
    if (q > 0) bias += alpha * aw * rsb[q - 1];
    if (q < Ss - 1) bias += beta * mw * rsb[q + 1];
    qbias[r] = bias;
    mrow[r] = -__builtin_inff();
    lrow[r] = 0.f;
  }

  // Q A-fragments for all 16 K-steps of the 64-deep head dim.
  const float* qrow = qbase + (size_t)(q0 + l16) * Dd;
  v2f qf[16];
#pragma unroll
  for (int dk = 0; dk < 16; ++dk) {
    qf[dk].x = qrow[dk * 4 + 2 * half];
    qf[dk].y = qrow[dk * 4 + 2 * half + 1];
  }

  constexpr int NT = Ss / 16;

  // ---------------- Pass 1: row max / sum (online) ----------------
  tdm_load_k16x64(kbase, kb_off[0]);
  for (int kt = 0; kt < NT; ++kt) {
    if (kt + 1 < NT) {
      tdm_load_k16x64(kbase + (size_t)(kt + 1) * 16 * Dd, kb_off[(kt + 1) & 1]);
      __builtin_amdgcn_s_wait_tensorcnt(1);
    } else {
      __builtin_amdgcn_s_wait_tensorcnt(0);
    }
    const float* kl = &kbuf[kt & 1][0];
    v8f acc = {};
#pragma unroll
    for (int dk = 0; dk < 16; ++dk) {
      v2f kf;
      kf.x = kl[l16 * 65 + dk * 4 + 2 * half];
      kf.y = kl[l16 * 65 + dk * 4 + 2 * half + 1];
      acc = wmma_f32(qf[dk], kf, acc);
    }
#pragma unroll
    for (int r = 0; r < 8; ++r) {
      const int q = q0 + r + 8 * half;
      float lg = acc[r] * scale +
                 mask[(size_t)q * Ss + kt * 16 + l16] * (-1e9f) + qbias[r];
      float t = lg;
      t = fmaxf(t, __shfl_xor(t, 1, 32));
      t = fmaxf(t, __shfl_xor(t, 2, 32));
      t = fmaxf(t, __shfl_xor(t, 4, 32));
      t = fmaxf(t, __shfl_xor(t, 8, 32));
      const float mnew = fmaxf(mrow[r], t);
      float e = __expf(lg - mnew);
      e += __shfl_xor(e, 1, 32);
      e += __shfl_xor(e, 2, 32);
      e += __shfl_xor(e, 4, 32);
      e += __shfl_xor(e, 8, 32);
      lrow[r] = lrow[r] * __expf(mrow[r] - mnew) + e;
      mrow[r] = mnew;
    }
  }
#pragma unroll
  for (int r = 0; r < 8; ++r) lrow[r] = 1.0f / lrow[r];

  // ---------------- Pass 2: write attn once, ctx = P @ V ----------------
  v8f cacc[4] = {v8f{}, v8f{}, v8f{}, v8f{}};
  float* attnb = attn + (size_t)bh * Ss * Ss;
  tdm_load_k16x64(kbase, kb_off[0]);
  for (int kt = 0; kt < NT; ++kt) {
    if (kt + 1 < NT) {
      tdm_load_k16x64(kbase + (size_t)(kt + 1) * 16 * Dd, kb_off[(kt + 1) & 1]);
      __builtin_amdgcn_s_wait_tensorcnt(1);
    } else {
      __builtin_amdgcn_s_wait_tensorcnt(0);
    }
    const float* kl = &kbuf[kt & 1][0];
    v8f acc = {};
#pragma unroll
    for (int dk = 0; dk < 16; ++dk) {
      v2f kf;
      kf.x = kl[l16 * 65 + dk * 4 + 2 * half];
      kf.y = kl[l16 * 65 + dk * 4 + 2 * half + 1];
      acc = wmma_f32(qf[dk], kf, acc);
    }
#pragma unroll
    for (int r = 0; r < 8; ++r) {
      const int q = q0 + r + 8 * half;
      float lg = acc[r] * scale +
                 mask[(size_t)q * Ss + kt * 16 + l16] * (-1e9f) + qbias[r];
      const float p = __expf(lg - mrow[r]) * lrow[r];
      attnb[(size_t)q * Ss + kt * 16 + l16] = p;  // single attn write
      pT[(r + 8 * half) * 16 + l16] = p;          // C-layout -> LDS
    }
    __syncthreads();
#pragma unroll
    for (int kk = 0; kk < 16; kk += 4) {
      v2f pa;  // A-layout fragment of P from LDS
      pa.x = pT[l16 * 16 + kk + 2 * half];
      pa.y = pT[l16 * 16 + kk + 2 * half + 1];
#pragma unroll
      for (int nt = 0; nt < 4; ++nt) {
        const float* vr =
            vbase + (size_t)(kt * 16 + kk + 2 * half) * Dd + nt * 16 + l16;
        v2f vb;
        vb.x = vr[0];
        vb.y = vr[Dd];
        cacc[nt] = wmma_f32(pa, vb, cacc[nt]);
      }
    }
    __syncthreads();
  }
#pragma unroll
  for (int nt = 0; nt < 4; ++nt)
#pragma unroll
    for (int r = 0; r < 8; ++r)
      ctx[((size_t)b * Ss + q0 + r + 8 * half) * Dd + h * DEPTH + nt * 16 +
          l16] = cacc[nt][r];
}

extern "C" void kernel_launch(void* const* d_in, const int* in_sizes, int n_in,
                              void* d_out, int out_size, void* d_ws,
                              size_t ws_size, hipStream_t stream) {
  const float* v = (const float*)d_in[0];
  const float* k = (const float*)d_in[1];
  const float* q = (const float*)d_in[2];
  const float* mask = (const float*)d_in[3];
  const float* wq = (const float*)d_in[4];
  const float* wqb = (const float*)d_in[5];
  const float* wk = (const float*)d_in[6];
  const float* wkb = (const float*)d_in[7];
  const float* wv = (const float*)d_in[8];
  const float* wvb = (const float*)d_in[9];
  const float* wo = (const float*)d_in[10];
  const float* wob = (const float*)d_in[11];
  const float* arw = (const float*)d_in[12];
  const float* maw = (const float*)d_in[13];
  const float* alp = (const float*)d_in[14];
  const float* bet = (const float*)d_in[15];

  const size_t BSD = (size_t)Bb * Ss * Dd;
  float* out = (float*)d_out;
  float* attn = out + BSD;  // outputs concatenated: out, attn
  float* ws = (float*)d_ws;
  float* Qp = ws;
  float* Kp = Qp + BSD;
  float* Vp = Kp + BSD;
  float* ctx = Vp + BSD;
  float* rsb = ctx + BSD;  // [B,H,S] head rowsums

  dim3 gblk(32);
  dim3 ggrid(Dd / 64, (Bb * Ss) / 16);
  gemm16x64<<<ggrid, gblk, 0, stream>>>(q, wq, wqb, Qp, Bb * Ss, Dd, Dd);
  gemm16x64<<<ggrid, gblk, 0, stream>>>(k, wk, wkb, Kp, Bb * Ss, Dd, Dd);
  gemm16x64<<<ggrid, gblk, 0, stream>>>(v, wv, wvb, Vp, Bb * Ss, Dd, Dd);
  rowsum_heads<<<dim3(Bb * Ss), dim3(512), 0, stream>>>(Vp, rsb);
  arma_attn<<<dim3(Ss / 16, Bb * Hh), dim3(32), 0, stream>>>(
      Qp, Kp, Vp, mask, rsb, arw, maw, alp, bet, attn, ctx);
  gemm16x64<<<ggrid, gblk, 0, stream>>>(ctx, wo, wob, out, Bb * Ss, Dd, Dd);
}